// SptPropagate_52948356825716
// MI455X (gfx1250) — compile-verified
//
#include <hip/hip_runtime.h>
#include <stdint.h>

#define TEMP 10.0f
#define BLK  256

// ---- CDNA5 async Global->LDS path (gfx1250), guarded so compile never breaks.
#if defined(__has_builtin)
#  if __has_builtin(__builtin_amdgcn_global_load_async_to_lds_b128)
#    define HAVE_ASYNC_B128 1
#  endif
#  if __has_builtin(__builtin_amdgcn_s_wait_asynccnt)
#    define WAIT_ASYNC(n) __builtin_amdgcn_s_wait_asynccnt(n)
#  endif
#endif
#if defined(HAVE_ASYNC_B128) && !defined(WAIT_ASYNC)
#  define WAIT_ASYNC(n) asm volatile("s_wait_asynccnt " #n ::: "memory")
#endif

#if defined(HAVE_ASYNC_B128)
typedef int v4i __attribute__((ext_vector_type(4)));

// Builtin signature (probed via diagnostics): (v4i AS1*, v4i AS3*, Ii offset, Ii cpol)
__device__ __forceinline__ void async_load_b128(const float4* g, float4* l)
{
    __builtin_amdgcn_global_load_async_to_lds_b128(
        (__attribute__((address_space(1))) v4i*)(v4i*)g,  // reinterpret + addrspacecast
        (__attribute__((address_space(3))) v4i*)(v4i*)l,  // reinterpret + generic->LDS cast
        0, 0);
}
#endif

// Streamed-once data: non-temporal loads (TH_LOAD_NT) keep the hot node/accum
// set resident in the 192MB L2 while 115MB of edge streams pass through.
template <typename T>
__device__ __forceinline__ T ntload(const T* p) { return __builtin_nontemporal_load(p); }

// ---------------------------------------------------------------------------
// Main edge pass (defined first so its disassembly is visible in feedback).
// edge_rel_q stream staged into LDS with GLOBAL_LOAD_ASYNC_TO_LDS_B128,
// double-buffered over grid-stride iterations. Each lane reads back only its
// own slot -> per-wave s_wait_asynccnt suffices, no workgroup barrier needed.
// ---------------------------------------------------------------------------
__global__ void __launch_bounds__(BLK) edge_accum_kernel(
    const float*  __restrict__ node_levels,
    const float4* __restrict__ node_q,
    const float4* __restrict__ edge_rel_q,
    const float*  __restrict__ edge_w,
    const int*    __restrict__ src,
    const int*    __restrict__ dst,
    const float*  __restrict__ m,
    float* __restrict__ Z,
    float* __restrict__ sum_q,    // [N][4]
    float* __restrict__ sum_l,
    int E)
{
#if defined(HAVE_ASYNC_B128)
    __shared__ float4 lbuf[2][BLK];   // 8 KB double buffer
#endif
    const int tid = threadIdx.x;
    const long long stride = (long long)gridDim.x * BLK;
    long long e = (long long)blockIdx.x * BLK + tid;

#if defined(HAVE_ASYNC_B128)
    int buf = 0;
    if (e < E) {
        async_load_b128(edge_rel_q + e, &lbuf[0][tid]);
    }
#endif

    for (; e < E; e += stride) {
        float4 a;  // rel quaternion (w,x,y,z) in (.x,.y,.z,.w)
#if defined(HAVE_ASYNC_B128)
        long long enext = e + stride;
        if (enext < E) {
            async_load_b128(edge_rel_q + enext, &lbuf[buf ^ 1][tid]);
            WAIT_ASYNC(1);   // oldest (current buf) complete, next in flight
        } else {
            WAIT_ASYNC(0);
        }
        a = lbuf[buf][tid];  // ds_load_b128
        buf ^= 1;
#else
        a = edge_rel_q[e];
#endif
        const int   s = ntload(src + e);
        const int   d = ntload(dst + e);
        const float w = ntload(edge_w + e);
        const float4 b = node_q[s];        // L2-resident gather
        const float  l = node_levels[s];   // L2-resident gather

        // Hamilton product rel_q * q_src
        float o0 = a.x*b.x - a.y*b.y - a.z*b.z - a.w*b.w;
        float o1 = a.x*b.y + a.y*b.x + a.z*b.w - a.w*b.z;
        float o2 = a.x*b.z - a.y*b.w + a.z*b.x + a.w*b.y;
        float o3 = a.x*b.w + a.y*b.z - a.z*b.y + a.w*b.x;

        float ex = __expf(TEMP * w * l - m[d]);

        atomicAdd(&Z[d],         ex);
        atomicAdd(&sum_l[d],     ex * l);
        atomicAdd(&sum_q[4*d+0], ex * o0);
        atomicAdd(&sum_q[4*d+1], ex * o1);
        atomicAdd(&sum_q[4*d+2], ex * o2);
        atomicAdd(&sum_q[4*d+3], ex * o3);
    }
}

// ---------------------------------------------------------------------------
// Seed m with self cost (handles empty segments == reference's
// max(seg_max, cost_self)), zero all accumulators in workspace.
// ws layout: sum_q[4N] | m[N] | Z[N] | sum_l[N]   (sum_q first => 16B aligned)
// ---------------------------------------------------------------------------
__global__ void __launch_bounds__(BLK) init_nodes_kernel(
    const float* __restrict__ node_levels,
    float4* __restrict__ sum_q, float* __restrict__ m,
    float* __restrict__ Z, float* __restrict__ sum_l, int N)
{
    int i = blockIdx.x * BLK + threadIdx.x;
    if (i >= N) return;
    m[i]     = TEMP * node_levels[i];   // cost_self, all >= 0
    Z[i]     = 0.0f;
    sum_l[i] = 0.0f;
    sum_q[i] = make_float4(0.0f, 0.0f, 0.0f, 0.0f);
}

// ---------------------------------------------------------------------------
// Segmented max via u32 atomic max on float bits (valid: costs >= 0).
// ---------------------------------------------------------------------------
__global__ void __launch_bounds__(BLK) edge_max_kernel(
    const float* __restrict__ node_levels,
    const float* __restrict__ edge_w,
    const int*   __restrict__ src,
    const int*   __restrict__ dst,
    unsigned int* __restrict__ m_bits, int E)
{
    int e = blockIdx.x * BLK + threadIdx.x;
    if (e >= E) return;
    float cost = TEMP * ntload(edge_w + e) * node_levels[ntload(src + e)];
    atomicMax(&m_bits[ntload(dst + e)], __float_as_uint(cost));
}

// ---------------------------------------------------------------------------
// Add self term, divide by Z, normalize quaternion, write output:
// d_out = out_q[4N] | out_levels[N]
// ---------------------------------------------------------------------------
__global__ void __launch_bounds__(BLK) finalize_kernel(
    const float*  __restrict__ node_levels,
    const float4* __restrict__ node_q,
    const float*  __restrict__ m,
    const float*  __restrict__ Z,
    const float4* __restrict__ sum_q,
    const float*  __restrict__ sum_l,
    float4* __restrict__ out_q,
    float*  __restrict__ out_levels, int N)
{
    int i = blockIdx.x * BLK + threadIdx.x;
    if (i >= N) return;
    float  l  = node_levels[i];
    float  es = __expf(TEMP * l - m[i]);   // exp_self
    float  z  = Z[i] + es;
    float4 q  = node_q[i];
    float4 sq = sum_q[i];
    float q0 = (sq.x + es * q.x) / z;
    float q1 = (sq.y + es * q.y) / z;
    float q2 = (sq.z + es * q.z) / z;
    float q3 = (sq.w + es * q.w) / z;
    float nrm = sqrtf(q0*q0 + q1*q1 + q2*q2 + q3*q3);
    nrm = fmaxf(nrm, 1e-12f);
    float rinv = 1.0f / nrm;
    out_q[i] = make_float4(q0 * rinv, q1 * rinv, q2 * rinv, q3 * rinv);
    out_levels[i] = (sum_l[i] + es * l) / z;
}

// ---------------------------------------------------------------------------
extern "C" void kernel_launch(void* const* d_in, const int* in_sizes, int n_in,
                              void* d_out, int out_size, void* d_ws, size_t ws_size,
                              hipStream_t stream)
{
    const float* node_levels = (const float*)d_in[0];   // [N]
    const float* node_q      = (const float*)d_in[1];   // [N,4]
    const float* edge_rel_q  = (const float*)d_in[2];   // [E,4]
    const float* edge_w      = (const float*)d_in[3];   // [E]
    const int*   src         = (const int*)  d_in[4];   // [E]
    const int*   dst         = (const int*)  d_in[5];   // [E]
    const int N = in_sizes[0];
    const int E = in_sizes[3];

    // Workspace: sum_q[4N] | m[N] | Z[N] | sum_l[N]  (7N floats = 2.8 MB)
    float* ws    = (float*)d_ws;
    float* sum_q = ws;              // 16B-aligned (base)
    float* m     = ws + (size_t)4 * N;
    float* Zacc  = ws + (size_t)5 * N;
    float* sum_l = ws + (size_t)6 * N;

    float* out_q      = (float*)d_out;
    float* out_levels = (float*)d_out + (size_t)4 * N;

    const int nblk_nodes = (N + BLK - 1) / BLK;
    const int nblk_edges = (E + BLK - 1) / BLK;
    int nblk_accum = nblk_edges < 4096 ? nblk_edges : 4096; // grid-stride, ~3 iters

    init_nodes_kernel<<<nblk_nodes, BLK, 0, stream>>>(
        node_levels, (float4*)sum_q, m, Zacc, sum_l, N);

    edge_max_kernel<<<nblk_edges, BLK, 0, stream>>>(
        node_levels, edge_w, src, dst, (unsigned int*)m, E);

    edge_accum_kernel<<<nblk_accum, BLK, 0, stream>>>(
        node_levels, (const float4*)node_q, (const float4*)edge_rel_q,
        edge_w, src, dst, m, Zacc, sum_q, sum_l, E);

    finalize_kernel<<<nblk_nodes, BLK, 0, stream>>>(
        node_levels, (const float4*)node_q, m, Zacc,
        (const float4*)sum_q, sum_l, (float4*)out_q, out_levels, N);
}